// NeuronLevelModels_66554813219390
// MI455X (gfx1250) — compile-verified
//
#include <hip/hip_runtime.h>
#include <math.h>

// Problem constants (match the reference).
#define BB 1024
#define DD 2048
#define MM 32
#define HH 32

// LDS row stride in floats: 36*4=144 bytes -> 16B aligned rows, and
// gcd(36,64)=4 with cycle length 16 -> conflict-free for 16 consecutive rows.
#define LSTR 36

typedef __attribute__((ext_vector_type(2))) float v2f;
typedef __attribute__((ext_vector_type(8))) float v8f;

__global__ __launch_bounds__(256)
void NeuronLevelModels_66554813219390_kernel(const float* __restrict__ pre,
                                             const float* __restrict__ w1,
                                             const float* __restrict__ b1,
                                             const float* __restrict__ wout,
                                             const float* __restrict__ bout,
                                             float* __restrict__ out)
{
    __shared__ float spre[128 * LSTR];   // 128 batch rows x 32 m (padded)
    __shared__ float sw1 [HH  * LSTR];   // 32 h rows x 32 m (padded)
    __shared__ float sb1 [HH];
    __shared__ float swo [HH];
    __shared__ float sbo;

    const int d   = blockIdx.y;          // neuron index
    const int b0  = blockIdx.x * 128;    // batch tile base
    const int tid = threadIdx.x;

    // ---- Stage pre tile: 128 rows x 32 floats = 1024 float4, 4 per thread.
    // Threads t..t+7 cover one 128B row -> fully coalesced within each row.
    #pragma unroll
    for (int p = 0; p < 4; ++p) {
        int idx = p * 256 + tid;         // 0..1023
        int row = idx >> 3;
        int q   = idx & 7;
        float4 v = *(const float4*)(pre + ((size_t)(b0 + row) * DD + d) * MM + q * 4);
        *(float4*)&spre[row * LSTR + q * 4] = v;
    }
    // ---- Stage w1[d]: 32x32 floats = 256 float4, 1 per thread.
    {
        int row = tid >> 3;
        int q   = tid & 7;
        float4 v = *(const float4*)(w1 + ((size_t)d * HH + row) * MM + q * 4);
        *(float4*)&sw1[row * LSTR + q * 4] = v;
    }
    if (tid < HH)               sb1[tid]      = b1  [(size_t)d * HH + tid];
    else if (tid < 2 * HH)      swo[tid - HH] = wout[(size_t)d * HH + (tid - HH)];
    else if (tid == 2 * HH)     sbo           = bout[d];
    __syncthreads();

    const int wave = tid >> 5;           // 8 waves
    const int lane = tid & 31;           // wave32
    const int half = lane >> 4;          // lane group 0/1
    const int lr   = lane & 15;
    const int r0   = wave * 16;          // this wave's 16 batch rows in tile

    // ---- Layer 1: D = A(16x32) x B(32x32) via 8 chained f32 WMMAs (K=4).
    // A frag layout (16x4 f32): lanes 0-15 hold K=k+{0,1}, lanes 16-31 K=k+{2,3}.
    // B frag mirrors with N across lanes; B[k][n] = w1[d][n][k].
    v8f c0 = {};                         // h = 0..15
    v8f c1 = {};                         // h = 16..31
    #pragma unroll
    for (int s = 0; s < 8; ++s) {
        const int k = 4 * s + 2 * half;
        v2f a, bA, bB;
        a.x  = spre[(r0 + lr) * LSTR + k];
        a.y  = spre[(r0 + lr) * LSTR + k + 1];
        bA.x = sw1 [ lr        * LSTR + k];
        bA.y = sw1 [ lr        * LSTR + k + 1];
        bB.x = sw1 [(16 + lr)  * LSTR + k];
        bB.y = sw1 [(16 + lr)  * LSTR + k + 1];
        c0 = __builtin_amdgcn_wmma_f32_16x16x4_f32(false, a, false, bA,
                                                   (short)0, c0, false, false);
        c1 = __builtin_amdgcn_wmma_f32_16x16x4_f32(false, a, false, bB,
                                                   (short)0, c1, false, false);
    }

    // ---- Fused epilogue: bias + exact-erf GELU + layer-2 dot + tanh.
    // C/D layout: VGPR v -> row (r0 + v + 8*half), column h = lr (c0) / 16+lr (c1).
    const float blo = sb1[lr], bhi = sb1[16 + lr];
    const float wlo = swo[lr], whi = swo[16 + lr];
    float t[8];
    #pragma unroll
    for (int v = 0; v < 8; ++v) {
        float x0 = c0[v] + blo;
        float x1 = c1[v] + bhi;
        float g0 = 0.5f * x0 * (1.0f + erff(x0 * 0.70710678118654752f));
        float g1 = 0.5f * x1 * (1.0f + erff(x1 * 0.70710678118654752f));
        t[v] = g0 * wlo + g1 * whi;      // partial sum over this lane's two h's
    }
    // Butterfly-reduce across the 16 lanes of each half (h = 0..31 total).
    #pragma unroll
    for (int off = 1; off < 16; off <<= 1) {
        #pragma unroll
        for (int v = 0; v < 8; ++v)
            t[v] += __shfl_xor(t[v], off, 32);
    }
    const float bo = sbo;
    #pragma unroll
    for (int v = 0; v < 8; ++v) {
        if (lr == v) {
            const int row = b0 + r0 + 8 * half + v;
            out[(size_t)row * DD + d] = tanhf(t[v] + bo);
        }
    }
}

extern "C" void kernel_launch(void* const* d_in, const int* in_sizes, int n_in,
                              void* d_out, int out_size, void* d_ws, size_t ws_size,
                              hipStream_t stream) {
    const float* pre  = (const float*)d_in[0];   // (B, D, M)
    const float* w1   = (const float*)d_in[1];   // (D, H, M)
    const float* b1   = (const float*)d_in[2];   // (D, H)
    const float* wout = (const float*)d_in[3];   // (D, 1, H)
    const float* bout = (const float*)d_in[4];   // (D,)
    float* out = (float*)d_out;                  // (B, D)

    dim3 grid(BB / 128, DD);
    NeuronLevelModels_66554813219390_kernel<<<grid, 256, 0, stream>>>(
        pre, w1, b1, wout, bout, out);
    (void)in_sizes; (void)n_in; (void)out_size; (void)d_ws; (void)ws_size;
}